// Camera_49203145343738
// MI455X (gfx1250) — compile-verified
//
#include <hip/hip_runtime.h>
#include <stdint.h>

#define TILE 64
#define BREG 70          // Bayer staging region: TILE + 2*3
#define BSTR 72          // padded row stride
#define NREG 68          // noisy region: TILE + 2*2
#define NSTR 70
#define NTHREADS 256     // 8 wave32 waves

__device__ __forceinline__ int refl(int i, int n) {
    i = (i < 0) ? -i : i;                 // np.pad 'reflect': -1 -> 1
    return (i >= n) ? (2 * n - 2 - i) : i; // n -> n-2
}

__global__ __launch_bounds__(NTHREADS)
void camera_isp_kernel(const float* __restrict__ im,
                       const float* __restrict__ yp,
                       const float* __restrict__ noise,
                       float* __restrict__ out,
                       int H, int W)
{
    __shared__ float sB[BREG * BSTR];   // gathered Bayer (raw im values, [0,1])
    __shared__ float sN[NREG * NSTR];   // noisy tile (with 2-halo)
    __shared__ float sYP[51];           // 3 x 17 tone-curve knots

    const int tid = threadIdx.x;
    const int tx0 = blockIdx.x * TILE;
    const int ty0 = blockIdx.y * TILE;
    const int b   = blockIdx.z;

    if (tid < 51) sYP[tid] = yp[tid];

    const float* imb = im + (size_t)b * 3 * H * W;

    // ---------------- Stage A: async channel-select gather of Bayer into LDS ----
    // CMAP = [[1,2],[0,1]]  =>  c = 1 + (x&1) - (y&1)
    for (int idx = tid; idx < BREG * BREG; idx += NTHREADS) {
        int iy = idx / BREG;
        int ix = idx - iy * BREG;
        int ay = refl(ty0 - 3 + iy, H);
        int ax = refl(tx0 - 3 + ix, W);
        int c  = 1 + (ax & 1) - (ay & 1);
        const float* g = imb + ((size_t)c * H + ay) * W + ax;
        uint32_t lds = (uint32_t)(uintptr_t)&sB[iy * BSTR + ix]; // low 32 bits of generic ptr = LDS offset
        asm volatile("global_load_async_to_lds_b32 %0, %1, off"
                     :: "v"(lds), "v"(g) : "memory");
    }
#if __has_builtin(__builtin_amdgcn_s_wait_asynccnt)
    __builtin_amdgcn_s_wait_asynccnt(0);
#else
    asm volatile("s_wait_asynccnt 0" ::: "memory");
#endif
    __syncthreads();

    // ---------------- Stage B: blur -> tone curve -> +noise, into LDS -----------
    constexpr float BW0 = 0.91922438f;   // exp(-1/(2*0.16)) normalized 3-tap Gaussian
    constexpr float BW1 = 0.04038781f;
    const float* nsb = noise + (size_t)b * H * W;

    for (int idx = tid; idx < NREG * NREG; idx += NTHREADS) {
        int iy = idx / NREG;
        int ix = idx - iy * NREG;
        // halo entries are evaluated at reflected *actual* coords so the 5x5 conv
        // sees exactly the reference's reflect-padded noisy image
        int ay = refl(ty0 - 2 + iy, H);
        int ax = refl(tx0 - 2 + ix, W);
        int br = ay - ty0 + 3;
        int bc = ax - tx0 + 3;

        const float* p0 = &sB[(br - 1) * BSTR + (bc - 1)];
        const float* p1 = p0 + BSTR;
        const float* p2 = p1 + BSTR;
        float r0 = BW1 * (p0[0] + p0[2]) + BW0 * p0[1];
        float r1 = BW1 * (p1[0] + p1[2]) + BW0 * p1[1];
        float r2 = BW1 * (p2[0] + p2[2]) + BW0 * p2[1];
        float blurred = 255.0f * (BW1 * (r0 + r2) + BW0 * r1);

        // jnp.interp with uniform knots XP = linspace(0,255,17)
        float s  = blurred * (16.0f / 255.0f);
        int   i0 = (int)s;
        i0 = (i0 < 0) ? 0 : ((i0 > 15) ? 15 : i0);
        float t  = s - (float)i0;
        int   c  = 1 + (ax & 1) - (ay & 1);
        float y0 = sYP[c * 17 + i0];
        float y1 = sYP[c * 17 + i0 + 1];
        float lin = fmaf(t, y1 - y0, y0);

        sN[iy * NSTR + ix] = lin + nsb[(size_t)ay * W + ax];
    }
    __syncthreads();

    // ---------------- Stage C: sparse 5x5 stencils + parity combine + b128 stores
    float* outb = out + (size_t)b * 3 * H * W;
    const size_t plane = (size_t)H * W;

    for (int gidx = tid; gidx < (TILE * TILE / 4); gidx += NTHREADS) {
        int gy  = gidx >> 4;            // 16 float4-groups per 64-wide row
        int gx4 = (gidx & 15) << 2;
        int y   = ty0 + gy;
        int x0  = tx0 + gx4;            // even (tx0 multiple of 64, gx4 multiple of 4)
        int py  = y & 1;

        float R[4], G[4], Bc[4];
#pragma unroll
        for (int j = 0; j < 4; ++j) {
            const float* w = &sN[gy * NSTR + gx4 + j]; // w[ky*NSTR+kx], ky,kx in 0..4
            float w22 = w[2 * NSTR + 2];                            // raw (noisy center)
            float s1  = w[1*NSTR+2] + w[2*NSTR+1] + w[2*NSTR+3] + w[3*NSTR+2]; // dist-1 cross
            float sx  = w[1*NSTR+1] + w[1*NSTR+3] + w[3*NSTR+1] + w[3*NSTR+3]; // diagonals
            float ver2 = w[0*NSTR+2] + w[4*NSTR+2];                 // w02 + w42
            float hor2 = w[2*NSTR+0] + w[2*NSTR+4];                 // w20 + w24
            float sc   = ver2 + hor2;                               // dist-2 cross

            float g_f = 0.125f * (4.0f * w22 + 2.0f * s1 - sc);
            float h_f = 0.125f * (5.0f * w22 + 4.0f * (w[2*NSTR+1] + w[2*NSTR+3])
                                  - hor2 + 0.5f * ver2 - sx);
            float v_f = 0.125f * (5.0f * w22 + 4.0f * (w[1*NSTR+2] + w[3*NSTR+2])
                                  - ver2 + 0.5f * hor2 - sx);
            float d_f = 0.125f * (6.0f * w22 + 2.0f * sx - 1.5f * sc);

            int px = j & 1;
            float r, gg, bb;
            if (py == 0) {
                if (px == 0) { r = v_f; gg = w22; bb = h_f; }   // p00 (G pixel)
                else         { r = d_f; gg = g_f; bb = w22; }   // p01 (B pixel)
            } else {
                if (px == 0) { r = w22; gg = g_f; bb = d_f; }   // p10 (R pixel)
                else         { r = h_f; gg = w22; bb = v_f; }   // p11 (G pixel)
            }
            const float inv = 1.0f / 255.0f;
            R[j]  = fminf(fmaxf(r  * inv, 0.0f), 1.0f);
            G[j]  = fminf(fmaxf(gg * inv, 0.0f), 1.0f);
            Bc[j] = fminf(fmaxf(bb * inv, 0.0f), 1.0f);
        }

        size_t o = (size_t)y * W + x0;
        *reinterpret_cast<float4*>(outb + o)             = make_float4(R[0],  R[1],  R[2],  R[3]);
        *reinterpret_cast<float4*>(outb + plane + o)     = make_float4(G[0],  G[1],  G[2],  G[3]);
        *reinterpret_cast<float4*>(outb + 2 * plane + o) = make_float4(Bc[0], Bc[1], Bc[2], Bc[3]);
    }
}

extern "C" void kernel_launch(void* const* d_in, const int* in_sizes, int n_in,
                              void* d_out, int out_size, void* d_ws, size_t ws_size,
                              hipStream_t stream) {
    (void)in_sizes; (void)n_in; (void)out_size; (void)d_ws; (void)ws_size;
    const float* im    = (const float*)d_in[0];   // (32, 3, 512, 512)
    const float* yp    = (const float*)d_in[1];   // (3, 17)
    const float* noise = (const float*)d_in[2];   // (32, 1, 512, 512)
    float* out = (float*)d_out;                   // (32, 3, 512, 512)

    const int B = 32, H = 512, W = 512;
    dim3 grid(W / TILE, H / TILE, B);             // 8 x 8 x 32 = 2048 blocks
    camera_isp_kernel<<<grid, NTHREADS, 0, stream>>>(im, yp, noise, out, H, W);
}